// EmbedAlignELBO_24077586661495
// MI455X (gfx1250) — compile-verified
//
#include <hip/hip_runtime.h>
#include <hip/hip_bf16.h>

// Problem constants (match the reference).
#define B_    32
#define L_EN_ 32
#define V_EN_ 32000
#define L_FR_ 4096
#define E_    128

typedef float v2f __attribute__((ext_vector_type(2)));
typedef float v8f __attribute__((ext_vector_type(8)));

// Workspace layout (floats):
//   [0, 1024)     : fr-term partials, stored as -sum(log(mean))
//   [1024, 1088)  : KL partials (+)
//   [1088]        : en-term partial, stored as -log_en
#define FR_BLOCKS 1024
#define KL_BLOCKS 64
#define WS_FR     0
#define WS_KL     1024
#define WS_EN     1088
#define WS_TOTAL  1089

__device__ __forceinline__ float block_reduce_sum(float v, float* red) {
  const int tid = threadIdx.x;
  red[tid] = v;
  __syncthreads();
  for (int s = blockDim.x >> 1; s > 0; s >>= 1) {
    if (tid < s) red[tid] += red[tid + s];
    __syncthreads();
  }
  return red[0];
}

// ---------------------------------------------------------------------------
// French term: for each (b, f): log( (1/32) * sum_l decoded_fr[b, l, f] ).
// One wave owns one (b, 16-wide f tile). The L_EN=32 reduction is done as
// 8 chained V_WMMA_F32_16X16X4_F32 ops with A = ones(16x4) in exact f32:
//   D[m, n] = sum_k B[k, n]  for every m  (column sums).
// B layout (4x16, 2 VGPRs): VGPR0 -> K = 2*half, VGPR1 -> K = 2*half+1,
// columns n = lane & 15. A K-permutation here is harmless: A is all-ones,
// so the column sum is invariant under any K arrangement.
// Grid: 1024 blocks x 256 threads (8 waves) -> 8192 tiles = 32 b x 256 ftiles.
// EXEC is all-ones at every WMMA (no divergence before it), per ISA rules.
// ---------------------------------------------------------------------------
__global__ void __launch_bounds__(256)
fr_term_kernel(const float* __restrict__ fr, float* __restrict__ ws) {
  __shared__ float red[256];
  const int tid  = threadIdx.x;
  const int wave = tid >> 5;
  const int lane = tid & 31;
  const int n    = lane & 15;
  const int half = lane >> 4;

  const int tile = blockIdx.x * 8 + wave;   // [0, 8192)
  const int b    = tile >> 8;               // 256 f-tiles per batch
  const int f0   = (tile & 255) << 4;

  const float* base = fr + (size_t)b * L_EN_ * L_FR_ + f0 + n;

  v2f a = {1.0f, 1.0f};  // A = ones(16x4): VGPR0/1 hold K slots, all 1.0
  v8f c = {};            // f32 accumulator (16x16)

#pragma unroll
  for (int l0 = 0; l0 < L_EN_; l0 += 4) {
    v2f bb;
    bb.x = base[(size_t)(l0 + 2 * half)     * L_FR_];
    bb.y = base[(size_t)(l0 + 2 * half + 1) * L_FR_];
    if (l0 + 4 < L_EN_) {
      // gfx1250 global_prefetch_b8: pull next K-slab toward the WGP.
      __builtin_prefetch(&base[(size_t)(l0 + 4 + 2 * half) * L_FR_], 0, 1);
    }
    // (neg_a, A, neg_b, B, c_mod, C, reuse_a, reuse_b)
    c = __builtin_amdgcn_wmma_f32_16x16x4_f32(false, a, false, bb, (short)0, c,
                                              false, false);
  }

  // Every lane's c[0] is the column sum for its n (rows identical; the two
  // lane-halves duplicate each other) -> keep only half == 0.
  float contrib =
      (half == 0) ? -__builtin_logf(c[0] * (1.0f / (float)L_EN_)) : 0.0f;

  const float s = block_reduce_sum(contrib, red);
  if (tid == 0) ws[WS_FR + blockIdx.x] = s;
}

// ---------------------------------------------------------------------------
// KL term: sum over 131072 elems of -log(sigma) + 0.5*(sigma^2 + mu^2) - 0.5.
// ---------------------------------------------------------------------------
__global__ void __launch_bounds__(256)
kl_term_kernel(const float* __restrict__ mus, const float* __restrict__ sigmas,
               float* __restrict__ ws) {
  __shared__ float red[256];
  const int N = B_ * L_EN_ * E_;  // 131072
  float acc = 0.0f;
  for (int i = blockIdx.x * blockDim.x + threadIdx.x; i < N;
       i += gridDim.x * blockDim.x) {
    const float m = mus[i];
    const float sg = sigmas[i];
    acc += fmaf(0.5f, fmaf(sg, sg, m * m), -0.5f) - __builtin_logf(sg);
  }
  const float s = block_reduce_sum(acc, red);
  if (threadIdx.x == 0) ws[WS_KL + blockIdx.x] = s;
}

// ---------------------------------------------------------------------------
// English term: 1024 gathers from decoded_en — never stream the 131 MB tensor.
// ---------------------------------------------------------------------------
__global__ void __launch_bounds__(256)
en_term_kernel(const int* __restrict__ sen, const float* __restrict__ den,
               float* __restrict__ ws) {
  __shared__ float red[256];
  float acc = 0.0f;
  for (int i = threadIdx.x; i < B_ * L_EN_; i += blockDim.x) {
    const int idx = sen[i];
    acc += -__builtin_logf(den[(size_t)i * V_EN_ + idx]);
  }
  const float s = block_reduce_sum(acc, red);
  if (threadIdx.x == 0) ws[WS_EN] = s;
}

// ---------------------------------------------------------------------------
// Finalize: deterministic fixed-order sum of all partials -> scalar output.
// ---------------------------------------------------------------------------
__global__ void __launch_bounds__(256)
finalize_kernel(const float* __restrict__ ws, float* __restrict__ out) {
  __shared__ float red[256];
  float acc = 0.0f;
  for (int i = threadIdx.x; i < WS_TOTAL; i += blockDim.x) acc += ws[i];
  const float s = block_reduce_sum(acc, red);
  if (threadIdx.x == 0) out[0] = s;  // kl - log_en - log_fr
}

extern "C" void kernel_launch(void* const* d_in, const int* in_sizes, int n_in,
                              void* d_out, int out_size, void* d_ws,
                              size_t ws_size, hipStream_t stream) {
  (void)in_sizes; (void)n_in; (void)out_size; (void)ws_size;
  const int*   sentence_en = (const int*)d_in[0];
  // d_in[1] (sentence_fr) is unused: only its static length (4096) matters.
  const float* mus        = (const float*)d_in[2];
  const float* sigmas     = (const float*)d_in[3];
  const float* decoded_en = (const float*)d_in[4];
  const float* decoded_fr = (const float*)d_in[5];
  float*       out        = (float*)d_out;
  float*       ws         = (float*)d_ws;

  fr_term_kernel<<<FR_BLOCKS, 256, 0, stream>>>(decoded_fr, ws);
  kl_term_kernel<<<KL_BLOCKS, 256, 0, stream>>>(mus, sigmas, ws);
  en_term_kernel<<<1, 256, 0, stream>>>(sentence_en, decoded_en, ws);
  finalize_kernel<<<1, 256, 0, stream>>>(ws, out);
}